// SimpleConvGCN_33835752358234
// MI455X (gfx1250) — compile-verified
//
#include <hip/hip_runtime.h>
#include <hip/hip_bf16.h>

typedef float v2f __attribute__((ext_vector_type(2)));
typedef float v8f __attribute__((ext_vector_type(8)));

#define NNODES 100000
#define FIN    512
#define H1C    16
#define H2C    32
#define NEDGE  3200000LL
#define NDDI   4096
#define DDI_IN 1056   // NUM_DRUGS + H2

// D = A(16x4 f32) * B(4x16 f32) + C (16x16 f32), exact fp32 on the matrix pipe.
__device__ __forceinline__ v8f wmma4(v2f a, v2f b, v8f c) {
  return __builtin_amdgcn_wmma_f32_16x16x4_f32(false, a, false, b, (short)0, c,
                                               false, false);
}

// ---------------- utility ----------------
__global__ void fill_f32(float* __restrict__ p, float v, long long n) {
  long long i = blockIdx.x * (long long)blockDim.x + threadIdx.x;
  long long stride = (long long)gridDim.x * blockDim.x;
  for (; i < n; i += stride) p[i] = v;
}

// ---------------- degree / norm ----------------
__global__ void degree_k(const long long* __restrict__ ei, float* __restrict__ deg) {
  long long e = blockIdx.x * (long long)blockDim.x + threadIdx.x;
  if (e >= NEDGE) return;
  int d = (int)ei[NEDGE + e];
  atomicAdd(&deg[d], 1.0f);
}

__global__ void rsqrt_k(float* __restrict__ deg) {
  int i = blockIdx.x * blockDim.x + threadIdx.x;
  if (i < NNODES) deg[i] = rsqrtf(deg[i]);
}

// ---------------- GEMM1: h1 = x @ W1  ([N,512]x[512,16]) ----------------
__global__ void gemm_x_w1(const float* __restrict__ x, const float* __restrict__ W1,
                          float* __restrict__ h1) {
  __shared__ float lw[FIN * H1C];  // 32 KB
  for (int i = threadIdx.x; i < FIN * H1C; i += blockDim.x) lw[i] = W1[i];
  __syncthreads();
  int lane = threadIdx.x & 31, wave = threadIdx.x >> 5;
  int tile = blockIdx.x * 8 + wave;               // 6250 row-tiles of 16
  if (tile >= NNODES / 16) return;                // whole wave exits together
  int m     = lane & 15;
  int khalf = (lane >> 4) * 2;                    // 0 or 2
  int col   = lane & 15;
  const float* xr = x + (size_t)(tile * 16 + m) * FIN;
  v8f acc = {0.f, 0.f, 0.f, 0.f, 0.f, 0.f, 0.f, 0.f};
#pragma unroll 8
  for (int k = 0; k < FIN; k += 4) {
    if ((k & 28) == 0)                            // one prefetch per 128B line group
      __builtin_prefetch(xr + k + 256, 0, 3);     // global_prefetch_b8, ~1KB ahead
    v2f a = *(const v2f*)(xr + k + khalf);        // lanes 0-15: K k..k+1, 16-31: k+2..k+3
    v2f b;
    b.x = lw[(k + khalf) * H1C + col];
    b.y = lw[(k + khalf + 1) * H1C + col];
    acc = wmma4(a, b, acc);
  }
  int rbase = tile * 16 + 8 * (lane >> 4);
  for (int r = 0; r < 8; ++r)
    h1[(size_t)(rbase + r) * H1C + col] = acc[r];
}

// ---------------- edge scatter (16-wide), L2-resident ----------------
__global__ void scatter16(const long long* __restrict__ ei, const float* __restrict__ h,
                          const float* __restrict__ dinv, float* __restrict__ agg) {
  long long t = blockIdx.x * (long long)blockDim.x + threadIdx.x;
  if (t >= NEDGE * 4) return;
  long long e = t >> 2;
  int chunk = (int)(t & 3);
  int s = (int)ei[e], d = (int)ei[NEDGE + e];
  float w = dinv[s] * dinv[d];
  float4 v = *(const float4*)(h + (size_t)s * H1C + chunk * 4);
  float* o = agg + (size_t)d * H1C + chunk * 4;
  atomicAdd(o + 0, v.x * w);
  atomicAdd(o + 1, v.y * w);
  atomicAdd(o + 2, v.z * w);
  atomicAdd(o + 3, v.w * w);
}

// ppi1 = relu(agg + h1 * dinv^2 + b1), written in place over agg
__global__ void finalize16(float* __restrict__ agg, const float* __restrict__ h,
                           const float* __restrict__ dinv, const float* __restrict__ b1) {
  int t = blockIdx.x * blockDim.x + threadIdx.x;
  if (t >= NNODES * H1C) return;
  int i = t >> 4, c = t & 15;
  float di = dinv[i];
  float v = agg[t] + h[t] * di * di + b1[c];
  agg[t] = fmaxf(v, 0.f);
}

// ---------------- GEMM2: h2 = ppi1 @ W2 ([N,16]x[16,32]) ----------------
__global__ void gemm_h_w2(const float* __restrict__ h, const float* __restrict__ W2,
                          float* __restrict__ h2) {
  __shared__ float lw[H1C * H2C];  // 2 KB
  for (int i = threadIdx.x; i < H1C * H2C; i += blockDim.x) lw[i] = W2[i];
  __syncthreads();
  int lane = threadIdx.x & 31, wave = threadIdx.x >> 5;
  int tile = blockIdx.x * 8 + wave;
  if (tile >= NNODES / 16) return;
  int m = lane & 15, khalf = (lane >> 4) * 2, col = lane & 15;
  const float* hr = h + (size_t)(tile * 16 + m) * H1C;
  v8f acc0 = {0.f, 0.f, 0.f, 0.f, 0.f, 0.f, 0.f, 0.f};
  v8f acc1 = {0.f, 0.f, 0.f, 0.f, 0.f, 0.f, 0.f, 0.f};
#pragma unroll
  for (int k = 0; k < H1C; k += 4) {
    v2f a = *(const v2f*)(hr + k + khalf);
    v2f b0, b1v;
    b0.x  = lw[(k + khalf) * H2C + col];
    b0.y  = lw[(k + khalf + 1) * H2C + col];
    b1v.x = lw[(k + khalf) * H2C + 16 + col];
    b1v.y = lw[(k + khalf + 1) * H2C + 16 + col];
    acc0 = wmma4(a, b0, acc0);
    acc1 = wmma4(a, b1v, acc1);
  }
  int rbase = tile * 16 + 8 * (lane >> 4);
  for (int r = 0; r < 8; ++r) {
    h2[(size_t)(rbase + r) * H2C + col]      = acc0[r];
    h2[(size_t)(rbase + r) * H2C + 16 + col] = acc1[r];
  }
}

// ---------------- edge scatter (32-wide) straight into d_out ----------------
__global__ void scatter32(const long long* __restrict__ ei, const float* __restrict__ h,
                          const float* __restrict__ dinv, float* __restrict__ agg) {
  long long t = blockIdx.x * (long long)blockDim.x + threadIdx.x;
  if (t >= NEDGE * 8) return;
  long long e = t >> 3;
  int chunk = (int)(t & 7);
  int s = (int)ei[e], d = (int)ei[NEDGE + e];
  float w = dinv[s] * dinv[d];
  float4 v = *(const float4*)(h + (size_t)s * H2C + chunk * 4);
  float* o = agg + (size_t)d * H2C + chunk * 4;
  atomicAdd(o + 0, v.x * w);
  atomicAdd(o + 1, v.y * w);
  atomicAdd(o + 2, v.z * w);
  atomicAdd(o + 3, v.w * w);
}

__global__ void finalize32(float* __restrict__ out, const float* __restrict__ h,
                           const float* __restrict__ dinv, const float* __restrict__ b2) {
  int t = blockIdx.x * blockDim.x + threadIdx.x;
  if (t >= NNODES * H2C) return;
  int i = t >> 5, c = t & 31;
  float di = dinv[i];
  float v = out[t] + h[t] * di * di + b2[c];
  out[t] = fmaxf(v, 0.f);
}

// ---------------- DDI fc1: relu([4096,1056] @ [1056,64] + b) ----------------
__global__ void ddi_fc1(const float* __restrict__ X, const float* __restrict__ W,
                        const float* __restrict__ bv, float* __restrict__ out) {
  int lane = threadIdx.x & 31, wave = threadIdx.x >> 5;
  int w = blockIdx.x * 8 + wave;   // 1024 waves total
  int mt = w >> 2, nt = w & 3;     // 256 row-tiles x 4 col-tiles
  int m = lane & 15, khalf = (lane >> 4) * 2;
  int col = nt * 16 + (lane & 15);
  const float* xr = X + (size_t)(mt * 16 + m) * DDI_IN;
  v8f acc = {0.f, 0.f, 0.f, 0.f, 0.f, 0.f, 0.f, 0.f};
#pragma unroll 8
  for (int k = 0; k < DDI_IN; k += 4) {
    if ((k & 28) == 0)
      __builtin_prefetch(xr + k + 256, 0, 3);
    v2f a = *(const v2f*)(xr + k + khalf);
    v2f b;
    b.x = W[(size_t)(k + khalf) * 64 + col];
    b.y = W[(size_t)(k + khalf + 1) * 64 + col];
    acc = wmma4(a, b, acc);
  }
  int rbase = mt * 16 + 8 * (lane >> 4);
  float bb = bv[col];
  for (int r = 0; r < 8; ++r)
    out[(size_t)(rbase + r) * 64 + col] = fmaxf(acc[r] + bb, 0.f);
}

// ---------------- DDI fc2: relu([4096,64] @ [64,16] + b) ----------------
__global__ void ddi_fc2(const float* __restrict__ X, const float* __restrict__ W,
                        const float* __restrict__ bv, float* __restrict__ out) {
  __shared__ float lw[64 * 16];
  for (int i = threadIdx.x; i < 64 * 16; i += blockDim.x) lw[i] = W[i];
  __syncthreads();
  int lane = threadIdx.x & 31, wave = threadIdx.x >> 5;
  int mt = blockIdx.x * 8 + wave;  // 256 row-tiles
  int m = lane & 15, khalf = (lane >> 4) * 2, col = lane & 15;
  const float* xr = X + (size_t)(mt * 16 + m) * 64;
  v8f acc = {0.f, 0.f, 0.f, 0.f, 0.f, 0.f, 0.f, 0.f};
#pragma unroll
  for (int k = 0; k < 64; k += 4) {
    v2f a = *(const v2f*)(xr + k + khalf);
    v2f b;
    b.x = lw[(k + khalf) * 16 + col];
    b.y = lw[(k + khalf + 1) * 16 + col];
    acc = wmma4(a, b, acc);
  }
  int rbase = mt * 16 + 8 * (lane >> 4);
  float bb = bv[col];
  for (int r = 0; r < 8; ++r)
    out[(size_t)(rbase + r) * 16 + col] = fmaxf(acc[r] + bb, 0.f);
}

// ---------------- DDI fc3: relu([4096,16] @ [16,1] + b) ----------------
__global__ void ddi_fc3(const float* __restrict__ X, const float* __restrict__ W,
                        const float* __restrict__ bv, float* __restrict__ out) {
  int r = blockIdx.x * blockDim.x + threadIdx.x;
  if (r >= NDDI) return;
  const float* xr = X + (size_t)r * 16;
  float s = bv[0];
#pragma unroll
  for (int j = 0; j < 16; ++j) s += xr[j] * W[j];
  out[r] = fmaxf(s, 0.f);
}

extern "C" void kernel_launch(void* const* d_in, const int* in_sizes, int n_in,
                              void* d_out, int out_size, void* d_ws, size_t ws_size,
                              hipStream_t stream) {
  const float*     x   = (const float*)d_in[0];
  const long long* ei  = (const long long*)d_in[1];
  const float*     ddi = (const float*)d_in[2];
  const float*     W1  = (const float*)d_in[3];
  const float*     b1  = (const float*)d_in[4];
  const float*     W2  = (const float*)d_in[5];
  const float*     b2  = (const float*)d_in[6];
  const float*     fW1 = (const float*)d_in[7];
  const float*     fb1 = (const float*)d_in[8];
  const float*     fW2 = (const float*)d_in[9];
  const float*     fb2 = (const float*)d_in[10];
  const float*     fW3 = (const float*)d_in[11];
  const float*     fb3 = (const float*)d_in[12];
  float* out = (float*)d_out;

  // workspace layout
  char*  ws   = (char*)d_ws;
  float* dinv = (float*)(ws);                                     // 100000 f32
  float* h1   = (float*)(ws + 0x80000);                           // N*16
  float* agg1 = (float*)(ws + 0x80000 + 0x640000);                // N*16 (becomes ppi1)
  float* h2   = (float*)(ws + 0x80000 + 2 * 0x640000);            // N*32
  float* t1   = (float*)(ws + 0x80000 + 2 * 0x640000 + 0xC80000); // 4096*64
  float* t2   = t1 + (size_t)NDDI * 64;                           // 4096*16
  float* out_ddi = out + (size_t)NNODES * H2C;

  // init: deg=1 (self-loop), agg1=0, layer-2 accumulator (d_out ppi region)=0
  fill_f32<<<2048, 256, 0, stream>>>(dinv, 1.0f, (long long)NNODES);
  fill_f32<<<2048, 256, 0, stream>>>(agg1, 0.0f, (long long)NNODES * H1C);
  fill_f32<<<2048, 256, 0, stream>>>(out, 0.0f, (long long)NNODES * H2C);

  // degrees -> dinv
  degree_k<<<(int)((NEDGE + 255) / 256), 256, 0, stream>>>(ei, dinv);
  rsqrt_k<<<(NNODES + 255) / 256, 256, 0, stream>>>(dinv);

  // layer 1
  gemm_x_w1<<<(NNODES / 16 + 7) / 8, 256, 0, stream>>>(x, W1, h1);
  scatter16<<<(int)((NEDGE * 4 + 255) / 256), 256, 0, stream>>>(ei, h1, dinv, agg1);
  finalize16<<<(NNODES * H1C + 255) / 256, 256, 0, stream>>>(agg1, h1, dinv, b1);

  // layer 2 (accumulate straight into d_out)
  gemm_h_w2<<<(NNODES / 16 + 7) / 8, 256, 0, stream>>>(agg1, W2, h2);
  scatter32<<<(int)((NEDGE * 8 + 255) / 256), 256, 0, stream>>>(ei, h2, dinv, out);
  finalize32<<<(NNODES * H2C + 255) / 256, 256, 0, stream>>>(out, h2, dinv, b2);

  // DDI branch
  ddi_fc1<<<128, 256, 0, stream>>>(ddi, fW1, fb1, t1);   // 1024 waves
  ddi_fc2<<<32, 256, 0, stream>>>(t1, fW2, fb2, t2);     // 256 waves
  ddi_fc3<<<(NDDI + 255) / 256, 256, 0, stream>>>(t2, fW3, fb3, out_ddi);
}